// GaussianSplatter_38534446579728
// MI455X (gfx1250) — compile-verified
//
#include <hip/hip_runtime.h>
#include <hip/hip_bf16.h>
#include <math.h>

typedef __attribute__((ext_vector_type(2))) float v2f;
typedef __attribute__((ext_vector_type(8))) float v8f;

#define HW   16384   // 128*128
#define NFC  48
#define C1C  6
#define PC   81
#define BC   32

// workspace layout (float offsets)
#define WS_PMSUM 0        // 81*64 = 5184
#define WS_KMAT  5184     // 64*64 = 4096
#define WS_GAP   9280     // 32*42 = 1344
#define WS_ATTN  10624    // 32*42 = 1344
#define WS_HR    11968    // 32*6*128*128 = 3145728

// ---------------------------------------------------------------------------
__global__ void k_zero(float* __restrict__ p, int n) {
  int i = blockIdx.x * 256 + threadIdx.x;
  if (i < n) p[i] = 0.f;
}

// ---------------------------------------------------------------------------
// Fused softmax over P=81 channels + mean-over-tiles binned reduction.
// grid = 32 batches * 8 (h%8 classes), block = 128 threads (one image row / iter).
__global__ void __launch_bounds__(128) k_softmax_pm(const float* __restrict__ logits,
                                                    float* __restrict__ pmsum) {
  int bi = blockIdx.x;
  int b = bi >> 3;
  int i = bi & 7;            // h % 8
  int tid = threadIdx.x;     // 0..127 == w
  int lane = tid & 31;
  __shared__ float acc[PC * 8];
  for (int e = tid; e < PC * 8; e += 128) acc[e] = 0.f;
  __syncthreads();
  int w = tid;
  for (int r = 0; r < 16; ++r) {
    int h = i + r * 8;
    const float* px = logits + (size_t)b * PC * HW + h * 128 + w;
    // pass 1: online max + sum of exp
    float mx = -3.4e38f, sm = 0.f;
    for (int p = 0; p < PC; ++p) {
      float x = px[p * HW];
      float nm = fmaxf(mx, x);
      sm = sm * __expf(mx - nm) + __expf(x - nm);
      mx = nm;
    }
    float inv = 1.f / sm;
    // pass 2: re-read (hot in WGP$/L2), reduce into position bins
    for (int p = 0; p < PC; ++p) {
      float pr = __expf(px[p * HW] - mx) * inv;
      pr += __shfl_xor(pr, 8, 32);
      pr += __shfl_xor(pr, 16, 32);
      if (lane < 8) atomicAdd(&acc[p * 8 + lane], pr);
    }
  }
  __syncthreads();
  for (int e = tid; e < PC * 8; e += 128) {
    int p = e >> 3, j = e & 7;
    atomicAdd(&pmsum[p * 64 + i * 8 + j], acc[e]);
  }
}

// ---------------------------------------------------------------------------
// Global average pool of lr_feat: one block per (b, c) of 42 channels.
__global__ void __launch_bounds__(256) k_gap(const float* __restrict__ feat,
                                             float* __restrict__ gap) {
  int idx = blockIdx.x;          // 0..1343
  int b = idx / 42, c = idx % 42;
  const float* p = feat + (size_t)(b * NFC + C1C + c) * HW;
  int tid = threadIdx.x;
  float s = 0.f;
  for (int e = tid; e < HW; e += 256) s += p[e];
  __shared__ float red[256];
  red[tid] = s; __syncthreads();
  for (int st = 128; st > 0; st >>= 1) {
    if (tid < st) red[tid] += red[tid + st];
    __syncthreads();
  }
  if (tid == 0) gap[idx] = red[0] * (1.f / 16384.f);
}

// ---------------------------------------------------------------------------
// Tiny: build the 64x64 K matrix (gaussian stamp bank) + 32x42 attn vector.
__global__ void __launch_bounds__(128) k_build(const float* __restrict__ pmsum,
                                               const float* __restrict__ sigx,
                                               const float* __restrict__ sigy,
                                               const float* __restrict__ opac,
                                               const float* __restrict__ rho,
                                               const float* __restrict__ gap,
                                               const float* __restrict__ wv2,
                                               float* __restrict__ Kmat,
                                               float* __restrict__ attn) {
  int tid = threadIdx.x;
  if (tid < 64) {
    int pos = tid;
    float sx = 0.f, sy = 0.f, op = 0.f, rr = 0.f;
    for (int p = 0; p < PC; ++p) {
      float pmv = pmsum[p * 64 + pos] * (1.f / 8192.f);
      sx += sigx[p] * pmv; sy += sigy[p] * pmv;
      op += opac[p] * pmv; rr += rho[p] * pmv;
    }
    float a = sx * sx, bc = rr * sx * sy, d = sy * sy;
    float det = a * d - bc * bc;
    float i00 = d / det, i01 = -bc / det, i11 = a / det;
    // kern = exp(z)/(2pi sqrt(det)); /max cancels the constant -> exp(z - zmax)
    float z[25]; float zmax = -3.4e38f;
    for (int u = 0; u < 5; ++u)
      for (int v = 0; v < 5; ++v) {
        float xx = -5.f + 2.5f * (float)u;
        float yy = -5.f + 2.5f * (float)v;
        float zz = -0.5f * (i00 * xx * xx + 2.f * i01 * xx * yy + i11 * yy * yy);
        z[u * 5 + v] = zz; zmax = fmaxf(zmax, zz);
      }
    int ic = pos >> 3, jc = pos & 7;
    for (int h = 0; h < 8; ++h)
      for (int w = 0; w < 8; ++w) {
        int dy = h - ic + 2, dx = w - jc + 2;
        float val = 0.f;
        if (dy >= 0 && dy < 5 && dx >= 0 && dx < 5)
          val = __expf(z[dy * 5 + dx] - zmax) * op;
        Kmat[pos * 64 + h * 8 + w] = val;
      }
  }
  // attn = sigmoid(w_v2 @ gap)
  for (int e = tid; e < BC * 42; e += 128) {
    int b = e / 42, o = e % 42;
    float s = 0.f;
    for (int ic2 = 0; ic2 < 42; ++ic2) s += wv2[o * 42 + ic2] * gap[b * 42 + ic2];
    attn[e] = 1.f / (1.f + __expf(-s));
  }
}

// ---------------------------------------------------------------------------
// WMMA render. GEMM rows are ordered channel-major: m = c*8192 + n (n = tile id),
// so a 16-row WMMA tile never crosses a channel boundary (8192 % 16 == 0) and all
// row decodes are shifts/masks (no div-by-6). Each wave: 16 rows x 64 cols,
// V_WMMA_F32_16X16X4_F32, 4 accumulators, 16 K-steps.
// A layout (ISA 7.12.2, 32-bit A 16x4): lane<16 holds K=0,1; lane>=16 holds K=2,3.
// K matrix staged in LDS pair-interleaved (lk[(k>>1)*128 + col*2 + (k&1)]) so each
// lane's B operand is one aligned ds_load_b64 landing in an even VGPR pair.
__global__ void __launch_bounds__(256) k_render(const float* __restrict__ feat,
                                                const float* __restrict__ V,
                                                const float* __restrict__ Kmat,
                                                float* __restrict__ hr) {
  __shared__ __align__(16) float lk[64 * 64];
  int tid = threadIdx.x;
  for (int e = tid; e < 4096; e += 256) {
    int k = e >> 6, col = e & 63;
    lk[((k >> 1) * 64 + col) * 2 + (k & 1)] = Kmat[e];
  }
  __syncthreads();

  int wave = (blockIdx.x * 256 + tid) >> 5;   // 3072 waves total
  int lane = tid & 31;
  int half = lane >> 4;     // 0 or 1
  int mr   = lane & 15;

  int cS  = wave >> 9;              // channel 0..5, uniform per wave
  int n0  = (wave << 4) & 8191;     // base tile index within channel

  // A-row for this lane: n = n0 + mr, channel cS
  int nA = n0 + mr;
  int b = nA >> 8, t = nA & 255, th = t >> 4, tw = t & 15;
  const float* rowp = feat + ((size_t)(b * NFC + cS) * 128 + th * 8) * 128 + tw * 8;

  v8f acc[4];
  #pragma unroll
  for (int nt = 0; nt < 4; ++nt) acc[nt] = (v8f){0.f,0.f,0.f,0.f,0.f,0.f,0.f,0.f};

  #pragma unroll
  for (int kk = 0; kk < 16; ++kk) {
    int k0 = kk * 4 + half * 2;                 // this lane's K pair (always even)
    v2f a = *(const v2f*)(rowp + ((k0 >> 3) * 128) + (k0 & 7));
    const float* bbase = lk + k0 * 64;          // == (k0>>1)*128
    #pragma unroll
    for (int nt = 0; nt < 4; ++nt) {
      int col = nt * 16 + mr;
      v2f bv = *(const v2f*)(bbase + col * 2);  // {K[k0][col], K[k0+1][col]}
      acc[nt] = __builtin_amdgcn_wmma_f32_16x16x4_f32(
          false, a, false, bv, (short)0, acc[nt], false, false);
    }
  }

  // D layout: N = lane&15, M = r + half*8. hr = acc * V1.
  #pragma unroll
  for (int r = 0; r < 8; ++r) {
    int n2 = n0 + r + half * 8;                 // same channel cS
    int b2 = n2 >> 8, t2 = n2 & 255, th2 = t2 >> 4, tw2 = t2 & 15;
    size_t vbase = ((size_t)(b2 * NFC + cS) * 128 + th2 * 8) * 128 + tw2 * 8;
    size_t hbase = ((size_t)(b2 * C1C + cS) * 128 + th2 * 8) * 128 + tw2 * 8;
    #pragma unroll
    for (int nt = 0; nt < 4; ++nt) {
      int col = nt * 16 + mr;
      int off = ((col >> 3) * 128) + (col & 7);
      hr[hbase + off] = acc[nt][r] * V[vbase + off];
    }
  }
}

// ---------------------------------------------------------------------------
// out = b_res + w_res(1x1) @ feat + bilinear_upsample(ms)   (full overwrite)
__global__ void __launch_bounds__(256) k_res(const float* __restrict__ feat,
                                             const float* __restrict__ ms,
                                             const float* __restrict__ wres,
                                             const float* __restrict__ bres,
                                             float* __restrict__ out) {
  __shared__ float wr[4 * 48];
  int tid = threadIdx.x;
  if (tid < 192) wr[tid] = wres[tid];
  __syncthreads();
  int pix = blockIdx.x * 256 + tid;
  int b = pix >> 14, hw = pix & 16383;
  int h = hw >> 7, w = hw & 127;
  float a0 = bres[0], a1 = bres[1], a2 = bres[2], a3 = bres[3];
  const float* fp = feat + (size_t)b * NFC * HW + hw;
  for (int c = 0; c < NFC; ++c) {
    float fv = fp[c * HW];
    a0 += fv * wr[c]; a1 += fv * wr[48 + c];
    a2 += fv * wr[96 + c]; a3 += fv * wr[144 + c];
  }
  // bilinear 32 -> 128 (half-pixel centers, edge clamp)
  float sh = (h + 0.5f) * 0.25f - 0.5f;
  float sw = (w + 0.5f) * 0.25f - 0.5f;
  int h0 = (int)floorf(sh); float fh = sh - (float)h0;
  int w0 = (int)floorf(sw); float fw = sw - (float)w0;
  int h0c = h0 < 0 ? 0 : h0, h1c = h0 + 1 > 31 ? 31 : h0 + 1;
  int w0c = w0 < 0 ? 0 : w0, w1c = w0 + 1 > 31 ? 31 : w0 + 1;
  float wa = (1.f - fh) * (1.f - fw), wb = (1.f - fh) * fw;
  float wc = fh * (1.f - fw), wd = fh * fw;
  const float* mp = ms + (size_t)b * 4 * 1024;
  float acc[4] = {a0, a1, a2, a3};
  for (int o = 0; o < 4; ++o) {
    const float* q = mp + o * 1024;
    acc[o] += wa * q[h0c * 32 + w0c] + wb * q[h0c * 32 + w1c] +
              wc * q[h1c * 32 + w0c] + wd * q[h1c * 32 + w1c];
    out[(size_t)(b * 4 + o) * HW + hw] = acc[o];
  }
}

// ---------------------------------------------------------------------------
// 3x3 conv over fused out_feat (ch<6: hr ; ch>=6: feat*attn*V), accumulate into out.
// One block = 16x16 output tile; 48ch x 18x18 patch (62 KB) + weights (6.75 KB) in LDS.
__global__ void __launch_bounds__(256) k_conv(const float* __restrict__ feat,
                                              const float* __restrict__ V,
                                              const float* __restrict__ hr,
                                              const float* __restrict__ attn,
                                              const float* __restrict__ wfine,
                                              const float* __restrict__ bfine,
                                              float* __restrict__ out) {
  __shared__ float sf[48 * 324];
  __shared__ float swf[1728];
  int tid = threadIdx.x;
  int blk = blockIdx.x;
  int b = blk >> 6, t = blk & 63, ty = t >> 3, tx = t & 7;
  for (int e = tid; e < 1728; e += 256) swf[e] = wfine[e];
  int gy0 = ty * 16 - 1, gx0 = tx * 16 - 1;
  for (int e = tid; e < 48 * 324; e += 256) {
    int c = e / 324, rem = e % 324;
    int py = rem / 18, px = rem % 18;
    int gh = gy0 + py, gw = gx0 + px;
    float v = 0.f;
    if (gh >= 0 && gh < 128 && gw >= 0 && gw < 128) {
      if (c < C1C) {
        v = hr[(size_t)(b * C1C + c) * HW + gh * 128 + gw];
      } else {
        size_t gi = (size_t)(b * NFC + c) * HW + gh * 128 + gw;
        v = feat[gi] * attn[b * 42 + (c - C1C)] * V[gi];
      }
    }
    sf[e] = v;
  }
  __syncthreads();
  int oy = tid >> 4, ox = tid & 15;
  int h = ty * 16 + oy, w = tx * 16 + ox;
  float acc[4] = {bfine[0], bfine[1], bfine[2], bfine[3]};
  for (int c = 0; c < NFC; ++c) {
    const float* sp = sf + c * 324 + oy * 18 + ox;
    float v0 = sp[0],  v1 = sp[1],  v2 = sp[2];
    float v3 = sp[18], v4 = sp[19], v5 = sp[20];
    float v6 = sp[36], v7 = sp[37], v8 = sp[38];
    const float* wp = swf + c * 9;
    #pragma unroll
    for (int o = 0; o < 4; ++o) {
      const float* wo = wp + o * 432;   // (o*48 + c)*9
      acc[o] += v0 * wo[0] + v1 * wo[1] + v2 * wo[2] +
                v3 * wo[3] + v4 * wo[4] + v5 * wo[5] +
                v6 * wo[6] + v7 * wo[7] + v8 * wo[8];
    }
  }
  size_t oi = (size_t)(b * 4) * HW + h * 128 + w;
  #pragma unroll
  for (int o = 0; o < 4; ++o) out[oi + o * HW] += acc[o];
}

// ---------------------------------------------------------------------------
extern "C" void kernel_launch(void* const* d_in, const int* in_sizes, int n_in,
                              void* d_out, int out_size, void* d_ws, size_t ws_size,
                              hipStream_t stream) {
  const float* ms     = (const float*)d_in[0];
  const float* feat   = (const float*)d_in[1];
  const float* V      = (const float*)d_in[2];
  const float* logits = (const float*)d_in[3];
  const float* sigx   = (const float*)d_in[4];
  const float* sigy   = (const float*)d_in[5];
  const float* opac   = (const float*)d_in[6];
  const float* rho    = (const float*)d_in[7];
  const float* wv2    = (const float*)d_in[8];
  const float* wfine  = (const float*)d_in[9];
  const float* bfine  = (const float*)d_in[10];
  const float* wres   = (const float*)d_in[11];
  const float* bres   = (const float*)d_in[12];
  float* out = (float*)d_out;
  float* ws  = (float*)d_ws;

  float* pmsum = ws + WS_PMSUM;
  float* Kmat  = ws + WS_KMAT;
  float* gap   = ws + WS_GAP;
  float* attn  = ws + WS_ATTN;
  float* hr    = ws + WS_HR;

  k_zero<<<(5184 + 255) / 256, 256, 0, stream>>>(pmsum, 5184);
  k_softmax_pm<<<256, 128, 0, stream>>>(logits, pmsum);
  k_gap<<<32 * 42, 256, 0, stream>>>(feat, gap);
  k_build<<<1, 128, 0, stream>>>(pmsum, sigx, sigy, opac, rho, gap, wv2, Kmat, attn);
  k_render<<<384, 256, 0, stream>>>(feat, V, Kmat, hr);
  k_res<<<2048, 256, 0, stream>>>(feat, ms, wres, bres, out);
  k_conv<<<2048, 256, 0, stream>>>(feat, V, hr, attn, wfine, bfine, out);
}